// Decoder_33380485825250
// MI455X (gfx1250) — compile-verified
//
#include <hip/hip_runtime.h>
#include <hip/hip_bf16.h>
#include <math.h>

// Problem constants (from reference setup_inputs)
#define Nb   1024
#define Lb   2048
#define Hb   128
#define STEP 2

typedef __attribute__((ext_vector_type(2))) float v2f;
typedef __attribute__((ext_vector_type(8))) float v8f;

__device__ __forceinline__ float sigmoidf_(float x) {
    return 1.0f / (1.0f + __expf(-x));
}

// ---------------------------------------------------------------------------
// GRU cell + q projection. One block per batch row n. 128 threads.
// x == nullptr means decoder_input = ones (step 0).
// ---------------------------------------------------------------------------
__global__ __launch_bounds__(128)
void gru_q_kernel(const float* __restrict__ x,      // (N,H) or nullptr
                  const float* __restrict__ h_in,   // (N,H)
                  const float* __restrict__ W_ih,   // (3H,H)
                  const float* __restrict__ W_hh,   // (3H,H)
                  const float* __restrict__ b_ih,   // (3H)
                  const float* __restrict__ b_hh,   // (3H)
                  const float* __restrict__ Wq,     // (H,H)
                  float* __restrict__ h_out,        // (N,H)
                  float* __restrict__ q_out)        // (N,H)
{
    const int n = blockIdx.x;
    const int t = threadIdx.x;           // 0..127, owns output index o = t

    __shared__ float xs[Hb];
    __shared__ float hs[Hb];
    __shared__ float hn[Hb];

    xs[t] = x ? x[(size_t)n * Hb + t] : 1.0f;
    hs[t] = h_in[(size_t)n * Hb + t];
    __syncthreads();

    const int o = t;
    float rx = b_ih[o], zx = b_ih[Hb + o], nx = b_ih[2 * Hb + o];
    float rh = b_hh[o], zh = b_hh[Hb + o], nh = b_hh[2 * Hb + o];
    const float* wr  = W_ih + (size_t)o * Hb;
    const float* wz  = W_ih + (size_t)(Hb + o) * Hb;
    const float* wn  = W_ih + (size_t)(2 * Hb + o) * Hb;
    const float* vr  = W_hh + (size_t)o * Hb;
    const float* vz  = W_hh + (size_t)(Hb + o) * Hb;
    const float* vn  = W_hh + (size_t)(2 * Hb + o) * Hb;
#pragma unroll 8
    for (int h = 0; h < Hb; ++h) {
        const float xv = xs[h], hv = hs[h];
        rx = __builtin_fmaf(xv, wr[h], rx);
        zx = __builtin_fmaf(xv, wz[h], zx);
        nx = __builtin_fmaf(xv, wn[h], nx);
        rh = __builtin_fmaf(hv, vr[h], rh);
        zh = __builtin_fmaf(hv, vz[h], zh);
        nh = __builtin_fmaf(hv, vn[h], nh);
    }
    const float r  = sigmoidf_(rx + rh);
    const float z  = sigmoidf_(zx + zh);
    const float nn = tanhf(nx + r * nh);
    const float hv = (1.0f - z) * nn + z * hs[t];
    hn[t] = hv;
    h_out[(size_t)n * Hb + t] = hv;
    __syncthreads();

    // q = h' @ Wq^T
    float qv = 0.0f;
    const float* wq = Wq + (size_t)o * Hb;
#pragma unroll 8
    for (int h = 0; h < Hb; ++h) qv = __builtin_fmaf(hn[h], wq[h], qv);
    q_out[(size_t)n * Hb + o] = qv;
}

// ---------------------------------------------------------------------------
// Fused attention scoring pass (the bandwidth-dominant kernel).
// k[n,l,o] = sum_h eo[n,l,h] * Wk[o,h]   (recomputed, never stored to HBM)
// logits[n,l] = sum_o tanh(q[n,o] + k[n,l,o]) * w_score[o]
//
// Grid: (L/128, N). 256 threads = 8 waves. Each wave handles a 16-row l-tile
// and the full o=0..127 range via 8 o-tiles of v_wmma_f32_16x16x4_f32,
// K-looped 32x over H=128. Wk lives in LDS (64 KB).
//
// fp32 16x16x4 A layout (ISA 7.12.2): lane%16 = M row; VGPR pair carries
// K and K+1; lane-half (lane>=16) shifts K by +2. B mirrors with N=lane%16.
// C/D: VGPR r -> M = r + 8*(lane/16), N = lane%16.
// ---------------------------------------------------------------------------
__global__ __launch_bounds__(256)
void attn_kernel(const float* __restrict__ eo,      // (N,L,H)
                 const float* __restrict__ Wk,      // (H,H)
                 const float* __restrict__ q,       // (N,H)
                 const float* __restrict__ w_score, // (H)
                 float* __restrict__ logits)        // (N,L)
{
    __shared__ float wkLds[Hb * Hb];                // 64 KB

    const int tid = threadIdx.x;
    const int n   = blockIdx.y;

    // Cooperative load of Wk into LDS (L2-resident source).
    for (int i = tid; i < Hb * Hb; i += 256) wkLds[i] = Wk[i];
    __syncthreads();

    const int wave = tid >> 5;               // 0..7 -> l sub-tile
    const int lane = tid & 31;
    const int li   = lane & 15;              // N / M position within tile
    const int half = lane >> 4;              // K shift selector

    const int l0 = blockIdx.x * 128 + wave * 16;

    // Load A tile: rows l0..l0+15, all K=0..127, into 32 v2f regs per lane.
    v2f a[32];
    const float* arow = eo + ((size_t)n * Lb + (l0 + li)) * Hb + 2 * half;
#pragma unroll
    for (int t = 0; t < 32; ++t)
        a[t] = *(const v2f*)(arow + 4 * t);

    float partial[8];
#pragma unroll
    for (int r = 0; r < 8; ++r) partial[r] = 0.0f;

    for (int ot = 0; ot < 8; ++ot) {
        const int o0 = ot * 16;
        v8f acc = {0.f, 0.f, 0.f, 0.f, 0.f, 0.f, 0.f, 0.f};
        const float* brow = wkLds + (size_t)(o0 + li) * Hb + 2 * half;
#pragma unroll
        for (int t = 0; t < 32; ++t) {
            v2f b = *(const v2f*)(brow + 4 * t);
            acc = __builtin_amdgcn_wmma_f32_16x16x4_f32(
                false, a[t], false, b, (short)0, acc, false, false);
        }
        // Fuse tanh(q + k) * w_score on the accumulator tile.
        const int o  = o0 + li;              // this lane's column
        const float qv = q[(size_t)n * Hb + o];
        const float wv = w_score[o];
#pragma unroll
        for (int r = 0; r < 8; ++r)
            partial[r] += tanhf(qv + acc[r]) * wv;
    }

    // Row-reduce across the 16 lanes of each half (masks 1..8 stay in-half).
#pragma unroll
    for (int r = 0; r < 8; ++r) {
#pragma unroll
        for (int m = 1; m < 16; m <<= 1)
            partial[r] += __shfl_xor(partial[r], m, 32);
    }

    if (li == 0) {
#pragma unroll
        for (int r = 0; r < 8; ++r)
            logits[(size_t)n * Lb + (l0 + r + 8 * half)] = partial[r];
    }
}

// ---------------------------------------------------------------------------
// Softmax over L + argmax (first-max tie-break) + gather next decoder input.
// One block per n, 256 threads, L=2048 -> 8 elements/thread.
// ---------------------------------------------------------------------------
__global__ __launch_bounds__(256)
void softmax_argmax_kernel(const float* __restrict__ logits, // (N,L)
                           const float* __restrict__ enc_in, // (N,L,H)
                           float* __restrict__ out,          // (N,STEP,L), slot s
                           int s,
                           float* __restrict__ dec_next)     // (N,H)
{
    const int n   = blockIdx.x;
    const int tid = threadIdx.x;
    const float* row = logits + (size_t)n * Lb;

    __shared__ float smax[256];
    __shared__ int   sidx[256];
    __shared__ float ssum[256];

    float lmax = -1e30f;
    int   lidx = 0;
    for (int l = tid; l < Lb; l += 256) {
        const float v = row[l];
        if (v > lmax) { lmax = v; lidx = l; }
    }
    smax[tid] = lmax; sidx[tid] = lidx;
    __syncthreads();
    for (int st = 128; st > 0; st >>= 1) {
        if (tid < st) {
            const float ov = smax[tid + st];
            const int   oi = sidx[tid + st];
            if (ov > smax[tid] || (ov == smax[tid] && oi < sidx[tid])) {
                smax[tid] = ov; sidx[tid] = oi;
            }
        }
        __syncthreads();
    }
    const float mx  = smax[0];
    const int   idx = sidx[0];

    float lsum = 0.0f;
    for (int l = tid; l < Lb; l += 256) lsum += __expf(row[l] - mx);
    ssum[tid] = lsum;
    __syncthreads();
    for (int st = 128; st > 0; st >>= 1) {
        if (tid < st) ssum[tid] += ssum[tid + st];
        __syncthreads();
    }
    const float inv = 1.0f / ssum[0];

    float* orow = out + (size_t)n * (STEP * Lb) + (size_t)s * Lb;
    for (int l = tid; l < Lb; l += 256)
        orow[l] = __expf(row[l] - mx) * inv;

    // Gather decoder_input for the next step.
    if (tid < Hb)
        dec_next[(size_t)n * Hb + tid] =
            enc_in[((size_t)n * Lb + idx) * Hb + tid];
}

// ---------------------------------------------------------------------------
extern "C" void kernel_launch(void* const* d_in, const int* in_sizes, int n_in,
                              void* d_out, int out_size, void* d_ws, size_t ws_size,
                              hipStream_t stream) {
    const float* enc_in  = (const float*)d_in[0];  // encoder_input  (N,L,H)
    const float* enc_out = (const float*)d_in[1];  // encoder_output (N,L,H)
    const float* state0  = (const float*)d_in[2];  // (N,H)
    const float* W_ih    = (const float*)d_in[3];
    const float* W_hh    = (const float*)d_in[4];
    const float* b_ih    = (const float*)d_in[5];
    const float* b_hh    = (const float*)d_in[6];
    const float* Wq      = (const float*)d_in[7];
    const float* Wk      = (const float*)d_in[8];
    const float* w_score = (const float*)d_in[9];
    float* out = (float*)d_out;

    float* ws      = (float*)d_ws;
    float* ws_state  = ws;                          // N*H
    float* ws_q      = ws_state + (size_t)Nb * Hb;  // N*H
    float* ws_dec    = ws_q     + (size_t)Nb * Hb;  // N*H
    float* ws_logits = ws_dec   + (size_t)Nb * Hb;  // N*L

    for (int s = 0; s < STEP; ++s) {
        const float* x = (s == 0) ? nullptr : ws_dec;
        const float* h = (s == 0) ? state0  : ws_state;
        gru_q_kernel<<<Nb, 128, 0, stream>>>(x, h, W_ih, W_hh, b_ih, b_hh, Wq,
                                             ws_state, ws_q);
        attn_kernel<<<dim3(Lb / 128, Nb), 256, 0, stream>>>(
            enc_out, Wk, ws_q, w_score, ws_logits);
        softmax_argmax_kernel<<<Nb, 256, 0, stream>>>(
            ws_logits, enc_in, out, s, ws_dec);
    }
}